// MaxChamferDistance_10625749090595
// MI455X (gfx1250) — compile-verified
//
#include <hip/hip_runtime.h>
#include <math.h>

typedef __attribute__((ext_vector_type(2))) float v2f;
typedef __attribute__((ext_vector_type(8))) float v8f;

#define NPTS   4096
#define NTILES 256      // 4096 / 16
#define NBATCH 8

// dist[b*NPTS + i] = min_j ||P[b,i] - Q[b,j]||^2
//
// Trick: pad the K=3 dot product to K=4 with A[:,3] = 1, B[3,:] = -|y|^2/2 and
// seed the WMMA accumulator with C[m][n] = -|x|^2/2.  Then the WMMA output is
//   c = x.y - |y|^2/2 - |x|^2/2 = -d2/2
// so min_n d2 = -2 * max_n c and the whole distance epilogue reduces to one
// fmax per C register per tile.
//
// grid = NBATCH*32 blocks of 256 threads (8 waves); wave w of a block handles
// x-tile (blockIdx%32)*8 + w of batch blockIdx/32.
__global__ __launch_bounds__(256)
void chamfer_min_kernel(const float* __restrict__ P,
                        const float* __restrict__ Q,
                        float* __restrict__ dist) {
    __shared__ float ldsQ[NPTS * 3];              // 48 KB, one batch of Q

    const int b    = blockIdx.x >> 5;
    const int g    = blockIdx.x & 31;
    const int tid  = threadIdx.x;
    const int wave = tid >> 5;
    const int lane = tid & 31;
    const int halfsel = lane >> 4;                // 0: lanes 0-15, 1: lanes 16-31
    const int l15  = lane & 15;

    const float* __restrict__ Pb = P + (size_t)b * NPTS * 3;
    const float* __restrict__ Qb = Q + (size_t)b * NPTS * 3;

    // Stage this batch's Q cloud into LDS (thread t reads 12 contiguous bytes).
    for (int p = tid; p < NPTS; p += 256) {
        ldsQ[3 * p + 0] = Qb[3 * p + 0];
        ldsQ[3 * p + 1] = Qb[3 * p + 1];
        ldsQ[3 * p + 2] = Qb[3 * p + 2];
    }
    __syncthreads();

    const int xtile  = g * 8 + wave;
    const int rowIdx = xtile * 16 + l15;

    // A fragment: 16x4 fp32.  Lane L<16 holds A[M=L][K=0..1] = (x0, x1);
    // lane L>=16 holds A[M=L-16][K=2..3] = (x2, 1.0).
    const float p0 = Pb[3 * rowIdx + 0];
    const float p1 = Pb[3 * rowIdx + 1];
    const float p2 = Pb[3 * rowIdx + 2];
    v2f afrag;
    afrag.x = halfsel ? p2 : p0;
    afrag.y = halfsel ? 1.0f : p1;

    // Accumulator seed: C VGPR i holds row M = i + 8*halfsel -> -|x_M|^2 / 2.
    v8f acc0;
#pragma unroll
    for (int i = 0; i < 8; ++i) {
        const int q  = xtile * 16 + i + (halfsel ? 8 : 0);
        const float c0 = Pb[3 * q + 0];
        const float c1 = Pb[3 * q + 1];
        const float c2 = Pb[3 * q + 2];
        acc0[i] = -0.5f * (c0 * c0 + c1 * c1 + c2 * c2);
    }

    float maxv[8];
#pragma unroll
    for (int i = 0; i < 8; ++i) maxv[i] = -3.4e38f;

    for (int t = 0; t < NTILES; ++t) {
        const int col = t * 16 + l15;
        const float q0 = ldsQ[3 * col + 0];
        const float q1 = ldsQ[3 * col + 1];
        const float q2 = ldsQ[3 * col + 2];
        // B fragment 4x16: lane L<16 -> (y0, y1); lane L>=16 -> (y2, -|y|^2/2)
        const float y2h = -0.5f * (q0 * q0 + q1 * q1 + q2 * q2);
        v2f bfrag;
        bfrag.x = halfsel ? q2 : q0;
        bfrag.y = halfsel ? y2h : q1;

        // c[m][n] = x_m . y_n - |y_n|^2/2 - |x_m|^2/2 = -d2(m,n)/2
        v8f c = __builtin_amdgcn_wmma_f32_16x16x4_f32(
            false, afrag, false, bfrag, (short)0, acc0, false, false);

#pragma unroll
        for (int i = 0; i < 8; ++i) maxv[i] = fmaxf(maxv[i], c[i]);
    }

    // Row-max across each 16-lane half (xor masks 1,2,4,8 stay within a half).
#pragma unroll
    for (int i = 0; i < 8; ++i) {
        float m = maxv[i];
        m = fmaxf(m, __shfl_xor(m, 1, 32));
        m = fmaxf(m, __shfl_xor(m, 2, 32));
        m = fmaxf(m, __shfl_xor(m, 4, 32));
        m = fmaxf(m, __shfl_xor(m, 8, 32));
        maxv[i] = m;
    }

    float* __restrict__ db = dist + (size_t)b * NPTS + xtile * 16;
    if (l15 == 0) {
#pragma unroll
        for (int i = 0; i < 8; ++i) db[i + halfsel * 8] = -2.0f * maxv[i];
    }
}

__device__ __forceinline__ float block_sum(float v, float* red, int tid) {
    red[tid] = v;
    __syncthreads();
    for (int s = 128; s > 0; s >>= 1) {
        if (tid < s) red[tid] += red[tid + s];
        __syncthreads();
    }
    const float r = red[0];
    __syncthreads();
    return r;
}

// One block per batch: Z-masked means of distA[b] and distB[b], summed.
__global__ __launch_bounds__(256)
void masked_mean_kernel(const float* __restrict__ distA,
                        const float* __restrict__ distB,
                        float* __restrict__ perBatch) {
    __shared__ float red[256];
    const int b   = blockIdx.x;
    const int tid = threadIdx.x;
    const float* __restrict__ A = distA + (size_t)b * NPTS;
    const float* __restrict__ B = distB + (size_t)b * NPTS;

    // pass 1: means
    float sA = 0.0f, sB = 0.0f;
    for (int i = tid; i < NPTS; i += 256) { sA += A[i]; sB += B[i]; }
    sA = block_sum(sA, red, tid);
    sB = block_sum(sB, red, tid);
    const float meanA = sA * (1.0f / NPTS);
    const float meanB = sB * (1.0f / NPTS);

    // pass 2: unbiased std (ddof=1) via deviations for accuracy
    float qA = 0.0f, qB = 0.0f;
    for (int i = tid; i < NPTS; i += 256) {
        float d = A[i] - meanA; qA += d * d;
        d = B[i] - meanB;       qB += d * d;
    }
    qA = block_sum(qA, red, tid);
    qB = block_sum(qB, red, tid);
    const float stdA = sqrtf(qA * (1.0f / (NPTS - 1)));
    const float stdB = sqrtf(qB * (1.0f / (NPTS - 1)));
    const float thrA = -0.5f * stdA;   // keep where (v - mean) > -z*std, z = 0.5
    const float thrB = -0.5f * stdB;

    // pass 3: masked sums and counts
    float mA = 0.0f, cA = 0.0f, mB = 0.0f, cB = 0.0f;
    for (int i = tid; i < NPTS; i += 256) {
        const float va = A[i];
        if ((va - meanA) > thrA) { mA += va; cA += 1.0f; }
        const float vb = B[i];
        if ((vb - meanB) > thrB) { mB += vb; cB += 1.0f; }
    }
    mA = block_sum(mA, red, tid);
    cA = block_sum(cA, red, tid);
    mB = block_sum(mB, red, tid);
    cB = block_sum(cB, red, tid);

    if (tid == 0) perBatch[b] = mA / cA + mB / cB;
}

__global__ void final_mean_kernel(const float* __restrict__ perBatch,
                                  float* __restrict__ out) {
    if (threadIdx.x == 0) {
        float s = 0.0f;
        for (int i = 0; i < NBATCH; ++i) s += perBatch[i];
        out[0] = s * (1.0f / NBATCH);
    }
}

extern "C" void kernel_launch(void* const* d_in, const int* in_sizes, int n_in,
                              void* d_out, int out_size, void* d_ws, size_t ws_size,
                              hipStream_t stream) {
    (void)in_sizes; (void)n_in; (void)out_size; (void)ws_size;
    const float* x = (const float*)d_in[0];   // [8, 4096, 3] fp32
    const float* y = (const float*)d_in[1];   // [8, 4096, 3] fp32
    float* ws       = (float*)d_ws;
    float* distA    = ws;                          // [8, 4096] min_m d2 (x -> y)
    float* distB    = ws + (size_t)NBATCH * NPTS;  // [8, 4096] min_n d2 (y -> x)
    float* perBatch = ws + (size_t)2 * NBATCH * NPTS;  // [8]

    chamfer_min_kernel<<<NBATCH * 32, 256, 0, stream>>>(x, y, distA);
    chamfer_min_kernel<<<NBATCH * 32, 256, 0, stream>>>(y, x, distB);
    masked_mean_kernel<<<NBATCH, 256, 0, stream>>>(distA, distB, perBatch);
    final_mean_kernel<<<1, 32, 0, stream>>>(perBatch, (float*)d_out);
}